// VtransE_70239895159019
// MI455X (gfx1250) — compile-verified
//
#include <hip/hip_runtime.h>
#include <stdint.h>

typedef __attribute__((ext_vector_type(2))) float v2f;
typedef __attribute__((ext_vector_type(4))) float v4f;
typedef __attribute__((ext_vector_type(8))) float v8f;

namespace {
constexpr int kB       = 1024;
constexpr int kFeat    = 25192;   // 100 + 4 + 512*49
constexpr int kNOut    = 70;
constexpr int kKSplit  = 4;
constexpr int kNChunk  = 514;     // 2 dense chunks (52 wide) + 512 channel chunks (49 -> pad 52)
constexpr int kKC      = 52;
constexpr int kThreads = 160;     // 5 waves, one per 16-wide N tile (70 -> 80)
}

// Fused: ROI-bilinear feature build (async global->LDS staging + LDS gather)
// + dual f32 WMMA GEMM (x_o*W_o^T accumulated with -x_s*W_s^T), split-K x4.
__global__ __launch_bounds__(kThreads)
void vtranse_fused(const float* __restrict__ subj, const float* __restrict__ obj,
                   const float* __restrict__ t_s,  const float* __restrict__ t_o,
                   const float* __restrict__ fm,   const float* __restrict__ scale,
                   const float* __restrict__ W_s,  const float* __restrict__ W_o,
                   const int* __restrict__ bbox_s, const int* __restrict__ bbox_o,
                   float* __restrict__ partial)
{
  __shared__ unsigned pidx[2][16][49];                         //  6,272 B
  __shared__ v4f      pw[2][16][49];                           // 25,088 B
  __shared__ __attribute__((aligned(16))) float fmbuf[16][196];// 12,544 B
  __shared__ __attribute__((aligned(16))) float feat[2][16][kKC]; // 6,656 B
  __shared__ float scl[kKC];                                   //    208 B

  const int tid = threadIdx.x;
  const int b0  = blockIdx.x * 16;   // 16-batch M tile
  const int ks  = blockIdx.y;        // split-K slice

  // ---- per-tile bilinear sampling parameters (align_corners 7x7 grid) ----
  for (int it = tid; it < 2 * 16 * 49; it += kThreads) {
    const int mat = it / 784;                 // 0 = object (+), 1 = subject (-)
    const int r   = it - mat * 784;
    const int m   = r / 49;
    const int p   = r - m * 49;
    const int* bb = (mat == 0 ? bbox_o : bbox_s) + 4 * (b0 + m);
    const int x0b = bb[0], y0b = bb[1], x1b = bb[2], y1b = bb[3];
    const int gy = p / 7;
    const int gx = p - gy * 7;
    const float ys = (float)y0b + (float)gy * (1.0f / 6.0f) * (float)(y1b - y0b - 1);
    const float xs = (float)x0b + (float)gx * (1.0f / 6.0f) * (float)(x1b - x0b - 1);
    const int y0i = (int)ys;                  // ys >= 0 -> trunc == floor
    const int x0i = (int)xs;
    int y1i = y0i + 1; if (y1i > y1b - 1) y1i = y1b - 1;
    int x1i = x0i + 1; if (x1i > x1b - 1) x1i = x1b - 1;
    const float wy = ys - (float)y0i;
    const float wx = xs - (float)x0i;
    const unsigned i00 = (unsigned)(y0i * 14 + x0i);
    const unsigned i01 = (unsigned)(y0i * 14 + x1i);
    const unsigned i10 = (unsigned)(y1i * 14 + x0i);
    const unsigned i11 = (unsigned)(y1i * 14 + x1i);
    pidx[mat][m][p] = i00 | (i01 << 8) | (i10 << 16) | (i11 << 24);
    pw[mat][m][p] = (v4f){(1.f - wy) * (1.f - wx), (1.f - wy) * wx,
                          wy * (1.f - wx),          wy * wx};
  }

  const int wave = tid >> 5;
  const int lane = tid & 31;
  const int half = lane >> 4;              // K/M half-split per WMMA layout
  const int ln   = lane & 15;
  const int n    = wave * 16 + ln;         // output column (up to 79; >=70 discarded)
  const unsigned na = (n < kNOut) ? (unsigned)n : (unsigned)(kNOut - 1);
  const int koff = half * 2;

  v8f acc = {};                            // 16x16 f32 accumulator (o*Wo + (-s)*Ws)

  for (int ch = ks; ch < kNChunk; ch += kKSplit) {
    const int kbase = (ch < 2) ? ch * kKC : 104 + (ch - 2) * 49;
    __syncthreads();                       // feat/fmbuf free (previous WMMAs done)

    if (tid < kKC) {                       // stage scale factors for this chunk
      int k = kbase + tid;
      if (k > kFeat - 1) k = kFeat - 1;
      scl[tid] = scale[k];
    }

    if (ch >= 2) {                         // async DMA: fm channel rows -> LDS
      const int c = ch - 2;
      for (int it = tid; it < 16 * 49; it += kThreads) {
        const int m = it / 49;
        const int q = it - m * 49;
        const float* g = fm + ((size_t)(b0 + m) * 512 + (size_t)c) * 196 + (size_t)q * 4;
        const unsigned lofs = (unsigned)(uintptr_t)&fmbuf[m][q * 4];
        const unsigned long long ga = (unsigned long long)(uintptr_t)g;
        asm volatile("global_load_async_to_lds_b128 %0, %1, off"
                     :: "v"(lofs), "v"(ga) : "memory");
      }
      asm volatile("s_wait_asynccnt 0x0" ::: "memory");
    }
    __syncthreads();

    // ---- feature staging: scaled (and sign-folded) A-matrix rows in LDS ----
    for (int it = tid; it < 2 * 16 * kKC; it += kThreads) {
      const int mat = it / (16 * kKC);
      const int r   = it - mat * (16 * kKC);
      const int m   = r / kKC;
      const int kc  = r - m * kKC;
      float v = 0.0f;
      if (ch >= 2) {
        if (kc < 49) {                     // kc in [49,52) stays zero padding
          const unsigned pk = pidx[mat][m][kc];
          const v4f w = pw[mat][m][kc];
          const float* row = &fmbuf[m][0];
          v = w.x * row[pk & 255u] + w.y * row[(pk >> 8) & 255u] +
              w.z * row[(pk >> 16) & 255u] + w.w * row[pk >> 24];
          v *= scl[kc];
        }
      } else {                             // dense head: subj/obj then t_s/t_o
        const int k = kbase + kc;          // < 104
        const int b = b0 + m;
        float raw;
        if (k < 100) raw = (mat == 0 ? obj : subj)[b * 100 + k];
        else         raw = (mat == 0 ? t_o : t_s)[b * 4 + (k - 100)];
        v = raw * scl[kc];
      }
      if (mat == 1) v = -v;                // fold the subtraction into A
      feat[mat][m][kc] = v;
    }
    __syncthreads();

    // prefetch next chunk's W rows (L2 hint; W stays L2-resident anyway)
    {
      const int chn = ch + kKSplit;
      if (chn < kNChunk && half == 0) {
        const int kb2 = (chn < 2) ? chn * kKC : 104 + (chn - 2) * 49;
        const size_t ofs = (size_t)na * kFeat + (size_t)kb2;
        __builtin_prefetch(&W_o[ofs], 0, 0);
        __builtin_prefetch(&W_s[ofs], 0, 0);
      }
    }

    // ---- WMMA over this K chunk: 13 steps of K=4, two matrices per step ----
#pragma unroll
    for (int st = 0; st < kKC / 4; ++st) {
      const int kc4 = st * 4;
      // A frag (16x4 f32): lane m=ln, k = kc4 + 2*half + {0,1}  (ds_load_b64)
      const v2f ao = *(const v2f*)&feat[0][ln][kc4 + koff];
      const v2f as = *(const v2f*)&feat[1][ln][kc4 + koff];
      // B frag (4x16 f32): column n, k = kc4 + 2*half + {0,1}
      unsigned ka = (unsigned)(kbase + kc4 + koff);
      if (ka > (unsigned)(kFeat - 2)) ka = (unsigned)(kFeat - 2); // pad clamp (A is 0 there)
      const size_t wofs = (size_t)na * (size_t)kFeat + ka;
      v2f bo, bs;
      bo.x = W_o[wofs]; bo.y = W_o[wofs + 1];
      bs.x = W_s[wofs]; bs.y = W_s[wofs + 1];
      acc = __builtin_amdgcn_wmma_f32_16x16x4_f32(false, ao, false, bo,
                                                  (short)0, acc, false, false);
      acc = __builtin_amdgcn_wmma_f32_16x16x4_f32(false, as, false, bs,
                                                  (short)0, acc, false, false);
    }
  }

  // ---- store split-K partial tile (disjoint regions, deterministic) ----
  if (n < kNOut) {
    float* dst = partial + (size_t)ks * (size_t)(kB * kNOut);
#pragma unroll
    for (int j = 0; j < 8; ++j) {
      const int m = j + 8 * half;          // C/D layout: VGPR j -> row j (+8 hi half)
      dst[(size_t)(b0 + m) * kNOut + n] = acc[j];
    }
  }
}

__global__ void reduce_ksplit(const float* __restrict__ partial,
                              float* __restrict__ out, int nelem)
{
  const int i = blockIdx.x * blockDim.x + threadIdx.x;
  if (i < nelem) {
    float s = partial[i];
#pragma unroll
    for (int k = 1; k < kKSplit; ++k) s += partial[(size_t)k * (size_t)nelem + i];
    out[i] = s;
  }
}

extern "C" void kernel_launch(void* const* d_in, const int* in_sizes, int n_in,
                              void* d_out, int out_size, void* d_ws, size_t ws_size,
                              hipStream_t stream)
{
  (void)in_sizes; (void)n_in; (void)out_size; (void)ws_size;
  const float* subj = (const float*)d_in[0];
  const float* obj  = (const float*)d_in[1];
  const float* t_s  = (const float*)d_in[2];
  const float* t_o  = (const float*)d_in[3];
  const float* fm   = (const float*)d_in[4];
  const float* scal = (const float*)d_in[5];
  const float* W_s  = (const float*)d_in[6];
  const float* W_o  = (const float*)d_in[7];
  const int*   bbs  = (const int*)d_in[8];
  const int*   bbo  = (const int*)d_in[9];
  float* out     = (float*)d_out;
  float* partial = (float*)d_ws;     // needs 4 * 1024 * 70 * 4 B ~= 1.1 MB

  dim3 grid(kB / 16, kKSplit);
  vtranse_fused<<<grid, kThreads, 0, stream>>>(subj, obj, t_s, t_o, fm, scal,
                                               W_s, W_o, bbs, bbo, partial);
  const int n = kB * kNOut;
  reduce_ksplit<<<(n + 255) / 256, 256, 0, stream>>>(partial, out, n);
}